// SphericalSmearing_86629490360985
// MI455X (gfx1250) — compile-verified
//
#include <hip/hip_runtime.h>
#include <math.h>

// Spherical harmonics (MAX_N=16) for 500k points.
// Store-bandwidth bound (512MB out @ 23.3TB/s ~ 22us). Strategy:
//  - 1 thread = 1 point, wave32 tile = 32 points = 32KB of output
//  - stage per-wave tile in padded LDS (<=2-way bank conflicts)
//  - drain LDS -> global with CDNA5 global_store_async_from_lds_b128
//    (ASYNCcnt path): 512B fully-coalesced global bytes per instruction.
// LDS addresses for the async ops are derived from the real smem pointer via
// an addrspace(3) ptrtoint, so the pointer escapes into the asm and the
// staging ds_store's (and all math feeding them) cannot be DCE'd.

#define MAXN   16
#define NPAIR  136
#define ROW_F  260          // padded LDS row stride in floats (256 + 4)
#define ROW_B  (ROW_F * 4)  // 1040 bytes
#define OUT_F  256
#define OUT_B  1024

struct Coeffs { float v[NPAIR]; };

// constexpr sqrt via Newton iteration (handles values down to ~1e-33)
constexpr double csqrt_(double x) {
  double g = x > 1.0 ? x : 1.0;
  for (int i = 0; i < 200; ++i) g = 0.5 * (g + x / g);
  return g;
}

constexpr Coeffs make_coeffs() {
  Coeffs c{};
  int idx = 0;
  for (int n = 0; n < MAXN; ++n) {
    for (int m = 0; m <= n; ++m) {
      double ratio = 1.0;                       // (n-m)! / (n+m)!
      for (int j = n - m + 1; j <= n + m; ++j) ratio /= (double)j;
      double val = (2.0 * n + 1.0) / (4.0 * 3.14159265358979323846) * ratio;
      c.v[idx++] = (float)csqrt_(val);
    }
  }
  return c;
}

constexpr Coeffs kC = make_coeffs();

// Generic -> LDS(addrspace 3) pointer, then to its 32-bit LDS byte offset.
// The ptrtoint captures the pointer, keeping LDS stores alive across the
// asm memory clobbers; the value is also the correct VSRC address for the
// async store-from-LDS instruction.
typedef __attribute__((address_space(3))) char lds_char;
__device__ __forceinline__ unsigned lds_addr_of(void* p) {
  return (unsigned)(unsigned long long)(lds_char*)p;
}

__global__ __launch_bounds__(32) void sph_harm_kernel(
    const float* __restrict__ xyz, float* __restrict__ out, int npts)
{
  __shared__ float smem[32 * ROW_F];   // 33,280 B
  const int lane = threadIdx.x;        // 0..31 (wave32)
  const int tile = blockIdx.x;         // tile of 32 points
  int pt  = tile * 32 + lane;
  int ptc = pt < npts ? pt : (npts - 1);

  float x = xyz[ptc * 3 + 0];
  float y = xyz[ptc * 3 + 1];
  float z = xyz[ptc * 3 + 2];

  float rinv = rsqrtf(x * x + y * y + z * z);
  float ct = z * rinv;
  float st = sqrtf(fmaxf(1.0f - ct * ct, 0.0f));
  float theta = atan2f(-y * rinv, -x * rinv) + 3.14159265358979323846f;
  float s1, c1;
  __sincosf(theta, &s1, &c1);

  float* row = &smem[lane * ROW_F];

  // Associated Legendre recurrences, fully unrolled; coeffs fold to literals.
  float pmm = 1.0f;                 // P[m][m]
  float cm = 1.0f, sm = 0.0f;       // cos(m*theta), sin(m*theta)
  #pragma unroll
  for (int m = 0; m < MAXN; ++m) {
    float p_nm   = pmm;             // P[n][m] at n=m
    float p_prev = 0.0f;            // P[n-1][m]
    #pragma unroll
    for (int n = m; n < MAXN; ++n) {
      float t = kC.v[n * (n + 1) / 2 + m] * p_nm;
      if (m == 0) {
        row[n] = t;                                 // out cols 0..15 (re, m=0)
      } else {
        int k = n * (n - 1) / 2 + (m - 1);
        row[16 + k]  = t * cm;                      // re, m>0: cols 16..135
        row[136 + k] = t * sm;                      // im, m>0: cols 136..255
      }
      if (n + 1 < MAXN) {
        // P[n+1][m] = ((2n+1) ct P[n][m] - (n+m) P[n-1][m]) / (n+1-m)
        float p_next = ((float)(2 * n + 1) * ct * p_nm
                        - (float)(n + m) * p_prev)
                       * (1.0f / (float)(n + 1 - m));
        p_prev = p_nm;
        p_nm   = p_next;
      }
    }
    pmm = -(float)(2 * m + 1) * st * pmm;           // P[m+1][m+1]
    float c_new = cm * c1 - sm * s1;                // angle recurrence
    float s_new = sm * c1 + cm * s1;
    cm = c_new; sm = s_new;
  }

  // LDS writes must land before the async engine reads them.
  asm volatile("s_wait_dscnt 0x0" ::: "memory");

  // Coalesced drain: 32 rows x 2 halves; each instruction stores
  // 32 lanes * 16B = 512B contiguous global bytes straight from LDS.
  const unsigned lds_base = lds_addr_of((void*)smem);
  const unsigned lane16   = (unsigned)(lane * 16);
  char* gbase = (char*)(out + (size_t)tile * 32 * OUT_F);
  int rows_valid = npts - tile * 32;

  if (rows_valid >= 32) {
    // Fast path (always taken for 500000 points): branch-free full tile.
    #pragma unroll
    for (int r = 0; r < 32; ++r) {
      #pragma unroll
      for (int h = 0; h < 2; ++h) {
        unsigned lds_addr = lds_base + (unsigned)(r * ROW_B + h * 512) + lane16;
        unsigned long long gaddr =
            (unsigned long long)(uintptr_t)(gbase + r * OUT_B + h * 512) + lane16;
        asm volatile("global_store_async_from_lds_b128 %0, %1, off"
                     :: "v"(gaddr), "v"(lds_addr) : "memory");
      }
    }
  } else {
    // Tail tile (unused for this problem size, kept for generality).
    for (int r = 0; r < rows_valid; ++r) {
      for (int h = 0; h < 2; ++h) {
        unsigned lds_addr = lds_base + (unsigned)(r * ROW_B + h * 512) + lane16;
        unsigned long long gaddr =
            (unsigned long long)(uintptr_t)(gbase + r * OUT_B + h * 512) + lane16;
        asm volatile("global_store_async_from_lds_b128 %0, %1, off"
                     :: "v"(gaddr), "v"(lds_addr) : "memory");
      }
    }
  }
  // Drain ASYNCcnt before the wave retires its LDS allocation.
  asm volatile("s_wait_asynccnt 0x0" ::: "memory");
}

extern "C" void kernel_launch(void* const* d_in, const int* in_sizes, int n_in,
                              void* d_out, int out_size, void* d_ws, size_t ws_size,
                              hipStream_t stream) {
  const float* xyz = (const float*)d_in[0];
  float* out = (float*)d_out;
  int npts = in_sizes[0] / 3;            // (500000, 3) flattened
  int blocks = (npts + 31) / 32;         // 500000/32 = 15625 exactly
  sph_harm_kernel<<<blocks, 32, 0, stream>>>(xyz, out, npts);
}